// HomoSAGE_56075093016767
// MI455X (gfx1250) — compile-verified
//
#include <hip/hip_runtime.h>

#define NN 100000
#define DD 128

typedef __attribute__((ext_vector_type(2))) float v2f;
typedef __attribute__((ext_vector_type(8))) float v8f;

// ---------------- zero workspace ----------------
__global__ void zero_f32_kernel(float* __restrict__ p, int n) {
    int i = blockIdx.x * blockDim.x + threadIdx.x;
    int stride = gridDim.x * blockDim.x;
    for (; i < n; i += stride) p[i] = 0.0f;
}

// ---------------- edge scatter-add (mean numerator + degree) ----------------
// One wave32 per edge (grid-strided): 4 floats per lane = 128 features.
// Hardware no-return f32 atomics; agg[] (51MB) is L2-resident on MI455X (192MB L2).
__global__ void __launch_bounds__(256)
sage_scatter_kernel(const float* __restrict__ h, const int* __restrict__ ei,
                    float* __restrict__ agg, float* __restrict__ deg, int E) {
    const int lane = threadIdx.x & 31;
    const int wid  = blockIdx.x * (blockDim.x >> 5) + (threadIdx.x >> 5);
    const int nw   = gridDim.x * (blockDim.x >> 5);
    for (int e = wid; e < E; e += nw) {
        const int src = __builtin_amdgcn_readfirstlane(ei[e]);
        const int dst = __builtin_amdgcn_readfirstlane(ei[E + e]);
        const float4 v = *reinterpret_cast<const float4*>(h + (size_t)src * DD + lane * 4);
        float* ap = agg + (size_t)dst * DD + lane * 4;
        unsafeAtomicAdd(ap + 0, v.x);
        unsafeAtomicAdd(ap + 1, v.y);
        unsafeAtomicAdd(ap + 2, v.z);
        unsafeAtomicAdd(ap + 3, v.w);
        if (lane == 0) unsafeAtomicAdd(deg + dst, 1.0f);
    }
}

// ---------------- fused SAGE linear: out = (agg/deg)@Wl^T + b + h@Wr^T ----------------
// One wave per 16x16 output tile. K=128 marched 4 at a time with
// V_WMMA_F32_16X16X4_F32 (exact fp32). Both GEMMs accumulate into one v8f C.
//
// WMMA f32 16x16x4 layouts (ISA 7.12.2):
//   A 16x4 : lane l -> M = l&15, VGPR0/1 = K = 2*(l>>4) + {0,1}
//   B 4x16 : lane l -> N = l&15, VGPR0/1 = K = 2*(l>>4) + {0,1}
//   C/D    : VGPR i -> M = i + 8*(l>>4), N = l&15
__global__ void __launch_bounds__(256)
sage_gemm_kernel(const float* __restrict__ agg, const float* __restrict__ deg,
                 const float* __restrict__ h,
                 const float* __restrict__ Wl, const float* __restrict__ Wr,
                 const float* __restrict__ bias, float* __restrict__ out, int relu) {
    const int lane  = threadIdx.x & 31;
    const int wid   = blockIdx.x * (blockDim.x >> 5) + (threadIdx.x >> 5);
    const int node0 = (wid >> 3) << 4;   // 6250 node tiles
    const int col0  = (wid & 7) << 4;    // 8 col tiles
    const int m     = lane & 15;
    const int half  = lane >> 4;
    const int row   = node0 + m;         // A-matrix row for this lane
    const int col   = col0 + m;          // B-matrix column for this lane

    const float s = 1.0f / fmaxf(deg[row], 1.0f);   // fused mean division

    const float2* aggp = reinterpret_cast<const float2*>(agg + (size_t)row * DD);
    const float2* hp   = reinterpret_cast<const float2*>(h   + (size_t)row * DD);
    // out = X @ W^T  =>  B[k][n] = W[n][k]  => row-major W rows are B columns
    const float2* wlp  = reinterpret_cast<const float2*>(Wl + (size_t)col * DD);
    const float2* wrp  = reinterpret_cast<const float2*>(Wr + (size_t)col * DD);

    v8f c = {0.f, 0.f, 0.f, 0.f, 0.f, 0.f, 0.f, 0.f};

#pragma unroll 8
    for (int k0 = 0; k0 < DD; k0 += 4) {
        const int kk = (k0 >> 1) + half;        // float2 index: element K = k0 + 2*half
        float2 a1f = aggp[kk];
        float2 a2f = hp[kk];
        float2 blf = wlp[kk];
        float2 brf = wrp[kk];
        v2f a1; a1.x = a1f.x * s; a1.y = a1f.y * s;
        v2f a2; a2.x = a2f.x;     a2.y = a2f.y;
        v2f bl; bl.x = blf.x;     bl.y = blf.y;
        v2f br; br.x = brf.x;     br.y = brf.y;
        c = __builtin_amdgcn_wmma_f32_16x16x4_f32(false, a1, false, bl, (short)0, c, false, false);
        c = __builtin_amdgcn_wmma_f32_16x16x4_f32(false, a2, false, br, (short)0, c, false, false);
    }

    const float bb = bias[col];                 // bias indexed by output column N
#pragma unroll
    for (int i = 0; i < 8; ++i) {
        float v = c[i] + bb;
        if (relu) v = fmaxf(v, 0.0f);
        out[(size_t)(node0 + i + 8 * half) * DD + col0 + m] = v;
    }
}

extern "C" void kernel_launch(void* const* d_in, const int* in_sizes, int n_in,
                              void* d_out, int out_size, void* d_ws, size_t ws_size,
                              hipStream_t stream) {
    const float* x   = (const float*)d_in[0];
    const int*   ei  = (const int*)  d_in[1];
    const float* W1l = (const float*)d_in[2];
    const float* W1r = (const float*)d_in[3];
    const float* b1  = (const float*)d_in[4];
    const float* W2l = (const float*)d_in[5];
    const float* W2r = (const float*)d_in[6];
    const float* b2  = (const float*)d_in[7];
    float* out = (float*)d_out;
    const int E = in_sizes[1] / 2;

    // workspace carve-up: agg[N*D] | deg[N] (padded) | h1[N*D]
    float* agg = (float*)d_ws;
    float* deg = agg + (size_t)NN * DD;
    float* h1  = deg + 131072;                  // keep h1 16B-aligned

    const int zeroN = NN * DD + NN;             // agg and deg are contiguous
    const dim3 zblk(256), zgrd(2048);
    const dim3 sblk(256), sgrd(4096);           // 32768 waves grid-striding E edges
    const dim3 gblk(256), ggrd(6250);           // 50000 waves = 6250x8 tiles

    // ---- layer 1 ----
    zero_f32_kernel<<<zgrd, zblk, 0, stream>>>(agg, zeroN);
    sage_scatter_kernel<<<sgrd, sblk, 0, stream>>>(x, ei, agg, deg, E);
    sage_gemm_kernel<<<ggrd, gblk, 0, stream>>>(agg, deg, x, W1l, W1r, b1, h1, 1);

    // ---- layer 2 ----
    zero_f32_kernel<<<zgrd, zblk, 0, stream>>>(agg, zeroN);
    sage_scatter_kernel<<<sgrd, sblk, 0, stream>>>(h1, ei, agg, deg, E);
    sage_gemm_kernel<<<ggrd, gblk, 0, stream>>>(agg, deg, h1, W2l, W2r, b2, out, 0);
}